// ACM3D_4733053960670
// MI455X (gfx1250) — compile-verified
//
#include <hip/hip_runtime.h>
#include <math.h>
#include <stdint.h>

#define NB 4
#define NC 512
#define NH 32
#define CPH 16
#define NSP 32768
#define NBLK1 16      // K1 blocks per (b,h): chunk of 2048 spatial
#define CH1 2048
#define NBLK3 8       // K3 blocks per (b,h): chunk of 4096 spatial
#define CH3 4096
#define NTILE (CH3 / 256)
#define TPAD 260      // LDS row pitch (floats): 260 % 64 == 4 -> conflict-free WMMA A reads

typedef float v2f __attribute__((ext_vector_type(2)));
typedef float v8f __attribute__((ext_vector_type(8)));

// ---- workspace layout (float offsets) ----
#define WS_SUMSP 0        // [128*16][16]  per-block channel sums
#define WS_BMK   32768    // [2048] per-block softmax max (K)
#define WS_BSK   34816    // [2048] per-block sumexp (K)
#define WS_BMQ   36864
#define WS_BSQ   38912
#define WS_MK    40960    // [128] global max (K)
#define WS_ISK   41088    // [128] 1/sumexp (K)
#define WS_MQ    41216
#define WS_ISQ   41344
#define WS_MU    41472    // [2048] mu[b*512+c]
#define WS_P     43520    // [2048] sigmoid gate
#define WS_D     45568    // [2048] (K-Q)*P
#define WS_KPART 47616    // [128*8][16]
#define WS_QPART 64000    // [128*8][16]

__device__ __forceinline__ void osm(float& m, float& s, float v) {
  float M = fmaxf(m, v);
  s = s * expf(m - M) + expf(v - M);
  m = M;
}
__device__ __forceinline__ void osm_pair(float& m, float& s, float om, float os) {
  float M = fmaxf(m, om);
  s = s * expf(m - M) + os * expf(om - M);
  m = M;
}

// async global -> LDS b128 (ASYNCcnt path), per-lane 16B
__device__ __forceinline__ void async_ld_b128(uint32_t lds_off, const float* gaddr) {
  asm volatile("global_load_async_to_lds_b128 %0, %1, off"
               :: "v"(lds_off), "v"(gaddr)
               : "memory");
}
__device__ __forceinline__ void wait_async0() {
  asm volatile("s_wait_asynccnt 0" ::: "memory");
}

// -------- K1: channel sums + online softmax stats (reads x once) --------
__global__ __launch_bounds__(256) void k1_stats(const float* __restrict__ x,
                                                const float* __restrict__ kw,
                                                const float* __restrict__ qw,
                                                float* __restrict__ ws) {
  __shared__ float s_cs[8][CPH];
  __shared__ float s_mk[8], s_sk[8], s_mq[8], s_sq[8];

  const int blk = blockIdx.x % NBLK1;
  const int bh  = blockIdx.x / NBLK1;
  const int h   = bh % NH;
  const int b   = bh / NH;
  const int t   = threadIdx.x;
  const int lane = t & 31, wave = t >> 5;
  const int rowbase = (b * NC + h * CPH) * NSP;
  const int nbase = blk * CH1;

  float kwr[CPH], qwr[CPH];
#pragma unroll
  for (int c = 0; c < CPH; ++c) { kwr[c] = kw[h*CPH+c]; qwr[c] = qw[h*CPH+c]; }

  float csum[CPH];
#pragma unroll
  for (int c = 0; c < CPH; ++c) csum[c] = 0.f;

  float mK = -INFINITY, sK = 0.f, mQ = -INFINITY, sQ = 0.f;

#pragma unroll
  for (int sweep = 0; sweep < 2; ++sweep) {
    const int n = nbase + sweep * 1024 + t * 4;
    float4 lk = make_float4(0.f, 0.f, 0.f, 0.f);
    float4 lq = make_float4(0.f, 0.f, 0.f, 0.f);
#pragma unroll
    for (int c = 0; c < CPH; ++c) {
      const float4 v = *(const float4*)(x + rowbase + c * NSP + n);
      csum[c] += (v.x + v.y) + (v.z + v.w);
      lk.x = fmaf(v.x, kwr[c], lk.x); lk.y = fmaf(v.y, kwr[c], lk.y);
      lk.z = fmaf(v.z, kwr[c], lk.z); lk.w = fmaf(v.w, kwr[c], lk.w);
      lq.x = fmaf(v.x, qwr[c], lq.x); lq.y = fmaf(v.y, qwr[c], lq.y);
      lq.z = fmaf(v.z, qwr[c], lq.z); lq.w = fmaf(v.w, qwr[c], lq.w);
    }
    osm(mK, sK, lk.x); osm(mK, sK, lk.y); osm(mK, sK, lk.z); osm(mK, sK, lk.w);
    osm(mQ, sQ, lq.x); osm(mQ, sQ, lq.y); osm(mQ, sQ, lq.z); osm(mQ, sQ, lq.w);
  }

  // wave32 reduce
#pragma unroll
  for (int off = 16; off > 0; off >>= 1) {
#pragma unroll
    for (int c = 0; c < CPH; ++c) csum[c] += __shfl_xor(csum[c], off, 32);
    osm_pair(mK, sK, __shfl_xor(mK, off, 32), __shfl_xor(sK, off, 32));
    osm_pair(mQ, sQ, __shfl_xor(mQ, off, 32), __shfl_xor(sQ, off, 32));
  }
  if (lane == 0) {
#pragma unroll
    for (int c = 0; c < CPH; ++c) s_cs[wave][c] = csum[c];
    s_mk[wave] = mK; s_sk[wave] = sK; s_mq[wave] = mQ; s_sq[wave] = sQ;
  }
  __syncthreads();
  if (t < CPH) {
    float s = 0.f;
#pragma unroll
    for (int w = 0; w < 8; ++w) s += s_cs[w][t];
    ws[WS_SUMSP + blockIdx.x * CPH + t] = s;
  }
  if (t == 0) {
    float m = s_mk[0], s = s_sk[0];
#pragma unroll
    for (int w = 1; w < 8; ++w) osm_pair(m, s, s_mk[w], s_sk[w]);
    ws[WS_BMK + blockIdx.x] = m; ws[WS_BSK + blockIdx.x] = s;
    m = s_mq[0]; s = s_sq[0];
#pragma unroll
    for (int w = 1; w < 8; ++w) osm_pair(m, s, s_mq[w], s_sq[w]);
    ws[WS_BMQ + blockIdx.x] = m; ws[WS_BSQ + blockIdx.x] = s;
  }
}

// -------- K2 (tiny): mu, global softmax stats, gate P --------
__global__ __launch_bounds__(256) void k2_mid(const float* __restrict__ pw1,
                                              const float* __restrict__ pb1,
                                              const float* __restrict__ pw2,
                                              const float* __restrict__ pb2,
                                              float* __restrict__ ws) {
  const int t = threadIdx.x;
  for (int i = t; i < NB * NC; i += 256) {
    const int b = i / NC, c = i % NC, h = c / CPH, cc = c % CPH;
    const int bh = b * NH + h;
    float s = 0.f;
    for (int blk = 0; blk < NBLK1; ++blk) s += ws[WS_SUMSP + (bh * NBLK1 + blk) * CPH + cc];
    ws[WS_MU + i] = s * (1.f / NSP);
  }
  for (int i = t; i < NB * NH; i += 256) {
    float m = -INFINITY, s = 0.f;
    for (int blk = 0; blk < NBLK1; ++blk)
      osm_pair(m, s, ws[WS_BMK + i * NBLK1 + blk], ws[WS_BSK + i * NBLK1 + blk]);
    ws[WS_MK + i] = m; ws[WS_ISK + i] = 1.f / s;
    m = -INFINITY; s = 0.f;
    for (int blk = 0; blk < NBLK1; ++blk)
      osm_pair(m, s, ws[WS_BMQ + i * NBLK1 + blk], ws[WS_BSQ + i * NBLK1 + blk]);
    ws[WS_MQ + i] = m; ws[WS_ISQ + i] = 1.f / s;
  }
  __syncthreads();
  // grouped bottleneck MLP on mu -> sigmoid gate P
  for (int g = t; g < NB * NH; g += 256) {
    const int b = g / NH, grp = g % NH;
    const float* muin = &ws[WS_MU + b * NC + grp * CPH];
    float h1[8];
#pragma unroll
    for (int j = 0; j < 8; ++j) {
      float a = pb1[grp * 8 + j];
#pragma unroll
      for (int i2 = 0; i2 < 16; ++i2) a = fmaf(pw1[(grp * 8 + j) * 16 + i2], muin[i2], a);
      h1[j] = fmaxf(a, 0.f);
    }
#pragma unroll
    for (int o = 0; o < 16; ++o) {
      float a = pb2[grp * 16 + o];
#pragma unroll
      for (int j = 0; j < 8; ++j) a = fmaf(pw2[(grp * 16 + o) * 8 + j], h1[j], a);
      ws[WS_P + b * NC + grp * CPH + o] = 1.f / (1.f + expf(-a));
    }
  }
}

// -------- K3: WMMA weighted pooling, async double-buffered staging --------
__global__ __launch_bounds__(256) void k3_pool(const float* __restrict__ x,
                                               const float* __restrict__ kw,
                                               const float* __restrict__ qw,
                                               float* __restrict__ ws) {
  __shared__ __align__(16) float xt[2][CPH][TPAD];
  __shared__ float aK[2][256], aQ[2][256];
  __shared__ float kacc[CPH], qacc[CPH];

  const int blk = blockIdx.x % NBLK3;
  const int bh  = blockIdx.x / NBLK3;
  const int h   = bh % NH;
  const int b   = bh / NH;
  const int t   = threadIdx.x;
  const int lane = t & 31, wave = t >> 5;
  const int rowbase = (b * NC + h * CPH) * NSP;

  const float MK = ws[WS_MK + bh], iSK = ws[WS_ISK + bh];
  const float MQ = ws[WS_MQ + bh], iSQ = ws[WS_ISQ + bh];

  float kwr[CPH], qwr[CPH];
#pragma unroll
  for (int c = 0; c < CPH; ++c) { kwr[c] = kw[h*CPH+c]; qwr[c] = qw[h*CPH+c]; }

  if (t < CPH) { kacc[t] = 0.f; qacc[t] = 0.f; }

  // per-thread staging coordinates (4 async b128 per tile)
  const int stc0 = t >> 6;          // channel sub-index 0..3
  const int stn4 = (t & 63) * 4;    // float offset in tile row
  const uint32_t lds0 = (uint32_t)(uintptr_t)&xt[0][0][0];
  const uint32_t lds1 = (uint32_t)(uintptr_t)&xt[1][0][0];

  // prologue: async-stage tile 0 into buffer 0
  {
    const int nb = blk * CH3;
#pragma unroll
    for (int cc = 0; cc < 4; ++cc) {
      const int c = cc * 4 + stc0;
      async_ld_b128(lds0 + (uint32_t)(c * TPAD + stn4) * 4u,
                    x + rowbase + c * NSP + nb + stn4);
    }
  }

  v8f acc = {0.f, 0.f, 0.f, 0.f, 0.f, 0.f, 0.f, 0.f};
  const int row  = lane & 15;   // A matrix M index
  const int hi   = lane >> 4;   // half-wave selector (K offset +2)
  const int ncol = lane & 15;   // B/D column
  const int krow = hi * 2;
  const bool isK = (ncol == 0);
  const bool isQ = (ncol == 1);

  for (int tile = 0; tile < NTILE; ++tile) {
    const int p = tile & 1;
    wait_async0();            // this wave's staged tile has landed in LDS
    __syncthreads();          // all waves' data visible; prev readers of buf p^1 done
    // prefetch next tile into the other buffer (overlaps with compute below)
    if (tile + 1 < NTILE) {
      const int nb = blk * CH3 + (tile + 1) * 256;
      const uint32_t ldsn = (p == 0) ? lds1 : lds0;
#pragma unroll
      for (int cc = 0; cc < 4; ++cc) {
        const int c = cc * 4 + stc0;
        async_ld_b128(ldsn + (uint32_t)(c * TPAD + stn4) * 4u,
                      x + rowbase + c * NSP + nb + stn4);
      }
    }
    // attention weights for this tile (recompute logits from LDS)
    {
      float lk = 0.f, lq = 0.f;
#pragma unroll
      for (int c = 0; c < CPH; ++c) {
        const float v = xt[p][c][t];
        lk = fmaf(v, kwr[c], lk);
        lq = fmaf(v, qwr[c], lq);
      }
      aK[p][t] = expf(lk - MK) * iSK;
      aQ[p][t] = expf(lq - MQ) * iSQ;
    }
    __syncthreads();
    // each wave: 32 n-positions, 8 chained WMMAs of K=4 (branchless B build)
#pragma unroll
    for (int kk = 0; kk < 8; ++kk) {
      const int base = wave * 32 + kk * 4;
      v2f A, Bm;
      const int acol = base + krow;
      A.x = xt[p][row][acol];
      A.y = xt[p][row][acol + 1];
      const float2 vk = *(const float2*)&aK[p][base + krow];
      const float2 vq = *(const float2*)&aQ[p][base + krow];
      Bm.x = isK ? vk.x : (isQ ? vq.x : 0.f);
      Bm.y = isK ? vk.y : (isQ ? vq.y : 0.f);
      acc = __builtin_amdgcn_wmma_f32_16x16x4_f32(false, A, false, Bm,
                                                  (short)0, acc, false, false);
    }
  }
  __syncthreads();
  // D column 0 = Ktilde (lanes 0/16), column 1 = Qtilde (lanes 1/17)
  if (ncol < 2) {
    float* dst = isK ? kacc : qacc;
#pragma unroll
    for (int r = 0; r < 8; ++r) atomicAdd(&dst[hi * 8 + r], acc[r]);
  }
  __syncthreads();
  if (t < CPH) {
    ws[WS_KPART + blockIdx.x * CPH + t] = kacc[t];
    ws[WS_QPART + blockIdx.x * CPH + t] = qacc[t];
  }
}

// -------- K4 (tiny): K,Q finalize, d=(K-Q)*P, dp --------
__global__ __launch_bounds__(256) void k4_finish(float* __restrict__ ws,
                                                 float* __restrict__ dp_out) {
  __shared__ float sdp[NB];
  const int t = threadIdx.x;
  if (t < NB) sdp[t] = 0.f;
  __syncthreads();
  for (int i = t; i < NB * NC; i += 256) {
    const int b = i / NC, c = i % NC, h = c / CPH, cc = c % CPH;
    const int bh = b * NH + h;
    float kt = 0.f, qt = 0.f;
    for (int blk = 0; blk < NBLK3; ++blk) {
      kt += ws[WS_KPART + (bh * NBLK3 + blk) * CPH + cc];
      qt += ws[WS_QPART + (bh * NBLK3 + blk) * CPH + cc];
    }
    const float muv = ws[WS_MU + i];
    const float K = kt - muv;
    const float Q = qt - muv;
    ws[WS_D + i] = (K - Q) * ws[WS_P + i];
    atomicAdd(&sdp[b], K * Q);
  }
  __syncthreads();
  if (t < NB) dp_out[t] = sdp[t] * (1.f / NC);
}

// -------- K5: y = P*x + (K-Q)*P (streaming) --------
__global__ __launch_bounds__(256) void k5_out(const float* __restrict__ x,
                                              const float* __restrict__ ws,
                                              float* __restrict__ y) {
  const int rowblk = blockIdx.x & 3;
  const int rc = blockIdx.x >> 2;            // b*512 + c
  const float a  = ws[WS_P + rc];
  const float dv = ws[WS_D + rc];
  const int base = rc * NSP + rowblk * (NSP / 4);
  const int t = threadIdx.x;
#pragma unroll
  for (int it = 0; it < 8; ++it) {
    const int o = base + (it * 256 + t) * 4;
    float4 v = *(const float4*)(x + o);
    v.x = fmaf(v.x, a, dv); v.y = fmaf(v.y, a, dv);
    v.z = fmaf(v.z, a, dv); v.w = fmaf(v.w, a, dv);
    *(float4*)(y + o) = v;
  }
}

extern "C" void kernel_launch(void* const* d_in, const int* in_sizes, int n_in,
                              void* d_out, int out_size, void* d_ws, size_t ws_size,
                              hipStream_t stream) {
  const float* x   = (const float*)d_in[0];
  const float* pw1 = (const float*)d_in[1];
  const float* pb1 = (const float*)d_in[2];
  const float* pw2 = (const float*)d_in[3];
  const float* pb2 = (const float*)d_in[4];
  const float* kw  = (const float*)d_in[5];
  // d_in[6] = kb, d_in[8] = qb: softmax is shift-invariant -> biases cancel.
  const float* qw  = (const float*)d_in[7];
  float* ws = (float*)d_ws;
  float* y  = (float*)d_out;
  float* dp = y + (size_t)NB * NC * NSP;

  k1_stats<<<NB * NH * NBLK1, 256, 0, stream>>>(x, kw, qw, ws);
  k2_mid<<<1, 256, 0, stream>>>(pw1, pb1, pw2, pb2, ws);
  k3_pool<<<NB * NH * NBLK3, 256, 0, stream>>>(x, kw, qw, ws);
  k4_finish<<<1, 256, 0, stream>>>(ws, dp);
  k5_out<<<NB * NC * 4, 256, 0, stream>>>(x, ws, y);
}